// BiLSTMLM_66159676227984
// MI455X (gfx1250) — compile-verified
//
#include <hip/hip_runtime.h>
#include <math.h>

// ---- problem constants (from reference) ----
#define VOCAB 32000
#define EMB   32
#define HID   16
#define SEQ   256
#define BATCH 16
#define GATES 64              // 4*HID
#define TWOH  32              // 2*HID
#define NTILES (VOCAB / 16)   // 2000 vocab tiles of 16
#define LWAVES 16             // waves per logits block (512 threads)
#define TPW   (NTILES / LWAVES) // 125 tiles per wave
#define CSHIFT 16.0f          // |logit| <= 32*0.25 + 0.25 = 8.25 < 16 -> exp(x-16) <= 1, no overflow

typedef __attribute__((ext_vector_type(2))) float v2f;
typedef __attribute__((ext_vector_type(8))) float v8f;

__device__ __forceinline__ float sigmoid_f(float x) {
  return 1.0f / (1.0f + __expf(-x));   // safe for all x (inf -> 0, 0 -> 1)
}
__device__ __forceinline__ float tanh_f(float x) {
  float t = __expf(-2.0f * fabsf(x));  // in (0,1], never overflows
  float r = (1.0f - t) / (1.0f + t);
  return copysignf(r, x);
}

// ---------------------------------------------------------------------------
// K1: embedding gather + input projection: xW[s,b,j] = (b_ih+b_hh)[j] + emb_row . W_ih[j,:]
// grid = SEQ*BATCH blocks, 64 threads (one gate output each)
// ---------------------------------------------------------------------------
__global__ __launch_bounds__(64) void bilstm_embed_xw(
    const int* __restrict__ idx, const float* __restrict__ emb,
    const float* __restrict__ W_ih, const float* __restrict__ b_ih,
    const float* __restrict__ b_hh, float* __restrict__ xW)
{
  __shared__ float x[EMB];
  const int sb = blockIdx.x;           // s*BATCH + b
  const int j  = threadIdx.x;          // gate index 0..63
  if (j < EMB) x[j] = emb[(size_t)idx[sb] * EMB + j];
  __syncthreads();
  float acc = b_ih[j] + b_hh[j];
#pragma unroll
  for (int e = 0; e < EMB; ++e) acc += x[e] * W_ih[j * EMB + e];
  xW[(size_t)sb * GATES + j] = acc;
}

// ---------------------------------------------------------------------------
// K2: the two sequential LSTM scans (shared cell). grid = 2 (dir), 512 threads.
// h,c,W_hh in LDS; per step only h@W_hh.T is on the critical path.
// Writes hid[s][b][dir*16 + j]  (hid[s] = concat(h_lr[s], h_rl[S-1-s]))
// ---------------------------------------------------------------------------
__global__ __launch_bounds__(512) void bilstm_scan(
    const float* __restrict__ xW, const float* __restrict__ W_hh,
    const float* __restrict__ h0_lr, const float* __restrict__ h0_rl,
    const float* __restrict__ c0_lr, const float* __restrict__ c0_rl,
    float* __restrict__ hid)
{
  __shared__ float h[BATCH][HID];
  __shared__ float c[BATCH][HID];
  __shared__ float g[BATCH][GATES];
  __shared__ float w[GATES][HID];

  const int dir = blockIdx.x;          // 0 = LR, 1 = RL
  const int tid = threadIdx.x;

  const float* h0 = dir ? h0_rl : h0_lr;
  const float* c0 = dir ? c0_rl : c0_lr;

  for (int i = tid; i < GATES * HID; i += 512) ((float*)w)[i] = W_hh[i];
  if (tid < BATCH * HID) {
    const int b = tid >> 4, j = tid & 15;
    const float hv = h0[tid];
    h[b][j] = hv;
    c[b][j] = c0[tid];
    const int srow = dir ? (SEQ - 1) : 0;          // h_lr[0]=h0 ; h_rl[0] pairs with s=S-1
    hid[srow * (BATCH * TWOH) + b * TWOH + dir * HID + j] = hv;
  }

  for (int t = 0; t < SEQ - 1; ++t) {
    __syncthreads();                               // h/c/w writes visible
    const int xt = dir ? (SEQ - 1 - t) : t;        // RL consumes emb[S-1], ..., emb[1]
    const float* xrow = xW + (size_t)xt * (BATCH * GATES);
#pragma unroll
    for (int q = tid; q < BATCH * GATES; q += 512) {
      const int b = q >> 6, j = q & 63;
      float acc = xrow[b * GATES + j];
#pragma unroll
      for (int k = 0; k < HID; ++k) acc += h[b][k] * w[j][k];
      g[b][j] = acc;
    }
    __syncthreads();
    if (tid < BATCH * HID) {
      const int b = tid >> 4, j = tid & 15;
      const float iv = sigmoid_f(g[b][j]);
      const float fv = sigmoid_f(g[b][j + 16]);
      const float gv = tanh_f   (g[b][j + 32]);
      const float ov = sigmoid_f(g[b][j + 48]);
      const float cv = fv * c[b][j] + iv * gv;
      const float hv = ov * tanh_f(cv);
      c[b][j] = cv;
      h[b][j] = hv;
      const int srow = dir ? (SEQ - 2 - t) : (t + 1); // h after t+1 steps
      hid[srow * (BATCH * TWOH) + b * TWOH + dir * HID + j] = hv;
    }
  }
}

// ---------------------------------------------------------------------------
// K3: logits = hid[s] (16x32) @ W_out.T + b_out, fused log_softmax.
// grid = SEQ blocks, 512 threads (16 waves, 125 vocab tiles of 16 each).
// fp32 WMMA 16x16x4, K=32 -> 8 chained WMMAs per tile.
// Pass A: sum(exp(x-16)) per row (fixed shift is overflow-safe given weight bounds).
// Pass B: recompute tiles (W_out is L2-hot) and NT-store x - lse.
// ---------------------------------------------------------------------------
__global__ __launch_bounds__(512) void bilstm_logits_lsm(
    const float* __restrict__ hid, const float* __restrict__ W_out,
    const float* __restrict__ b_out, float* __restrict__ out)
{
  __shared__ float part[LWAVES][BATCH];
  __shared__ float lse[BATCH];

  const int s    = blockIdx.x;
  const int tid  = threadIdx.x;
  const int wave = tid >> 5;
  const int lane = tid & 31;
  const int n    = lane & 15;       // N (vocab-in-tile) / M for A loads
  const int hi   = lane >> 4;       // lane half -> K offset 2*hi, D rows +8*hi

  // --- A operand: hid[s] 16x32, A[v] lane l: M=l%16, K=4*kc + v + 2*hi ---
  const float* hs = hid + (size_t)s * (BATCH * TWOH);
  v2f a[8];
#pragma unroll
  for (int kc = 0; kc < 8; ++kc)
    a[kc] = *(const v2f*)(hs + n * TWOH + 4 * kc + 2 * hi);

  // ---------------- pass A: row-wise sum of exp(x - CSHIFT) ----------------
  float psum[8];
#pragma unroll
  for (int r = 0; r < 8; ++r) psum[r] = 0.0f;

  for (int t = 0; t < TPW; ++t) {
    const int vbase = (wave * TPW + t) * 16;
    const float* wr = W_out + (size_t)(vbase + n) * TWOH + 2 * hi;
    v2f bt[8];
#pragma unroll
    for (int kc = 0; kc < 8; ++kc) bt[kc] = *(const v2f*)(wr + 4 * kc);
    const float bias = b_out[vbase + n];
    v8f acc;
#pragma unroll
    for (int r = 0; r < 8; ++r) acc[r] = bias;
#pragma unroll
    for (int kc = 0; kc < 8; ++kc)
      acc = __builtin_amdgcn_wmma_f32_16x16x4_f32(
          false, a[kc], false, bt[kc], (short)0, acc, false, false);
#pragma unroll
    for (int r = 0; r < 8; ++r) psum[r] += __expf(acc[r] - CSHIFT);
  }

  // reduce across the 16 lanes of each half (columns of this wave's tiles)
#pragma unroll
  for (int r = 0; r < 8; ++r) {
    float v = psum[r];
    v += __shfl_xor(v, 1, 16);
    v += __shfl_xor(v, 2, 16);
    v += __shfl_xor(v, 4, 16);
    v += __shfl_xor(v, 8, 16);
    psum[r] = v;
  }
  if ((lane & 15) == 0) {
#pragma unroll
    for (int r = 0; r < 8; ++r) part[wave][r + 8 * hi] = psum[r];
  }
  __syncthreads();
  if (tid < BATCH) {            // deterministic fixed-order combine across waves
    float ssum = 0.0f;
#pragma unroll
    for (int wv = 0; wv < LWAVES; ++wv) ssum += part[wv][tid];
    lse[tid] = CSHIFT + __logf(ssum);
  }
  __syncthreads();

  float lrow[8];
#pragma unroll
  for (int r = 0; r < 8; ++r) lrow[r] = lse[r + 8 * hi];

  // ---------------- pass B: recompute tiles, emit x - lse (NT stores) ------
  for (int t = 0; t < TPW; ++t) {
    const int vbase = (wave * TPW + t) * 16;
    const float* wr = W_out + (size_t)(vbase + n) * TWOH + 2 * hi;
    v2f bt[8];
#pragma unroll
    for (int kc = 0; kc < 8; ++kc) bt[kc] = *(const v2f*)(wr + 4 * kc);
    const float bias = b_out[vbase + n];
    v8f acc;
#pragma unroll
    for (int r = 0; r < 8; ++r) acc[r] = bias;
#pragma unroll
    for (int kc = 0; kc < 8; ++kc)
      acc = __builtin_amdgcn_wmma_f32_16x16x4_f32(
          false, a[kc], false, bt[kc], (short)0, acc, false, false);
    const size_t base = ((size_t)s * BATCH) * VOCAB + vbase + n;
#pragma unroll
    for (int r = 0; r < 8; ++r) {
      const int m = r + 8 * hi;   // batch row (D layout: VGPR r, lane half)
      __builtin_nontemporal_store(acc[r] - lrow[r], &out[base + (size_t)m * VOCAB]);
    }
  }
}

// ---------------------------------------------------------------------------
extern "C" void kernel_launch(void* const* d_in, const int* in_sizes, int n_in,
                              void* d_out, int out_size, void* d_ws, size_t ws_size,
                              hipStream_t stream) {
  const int*   idx   = (const int*)  d_in[0];   // [SEQ,BATCH] int32
  const float* emb   = (const float*)d_in[1];   // [VOCAB,EMB]
  const float* W_ih  = (const float*)d_in[2];   // [64,32]
  const float* W_hh  = (const float*)d_in[3];   // [64,16]
  const float* b_ih  = (const float*)d_in[4];   // [64]
  const float* b_hh  = (const float*)d_in[5];   // [64]
  const float* W_out = (const float*)d_in[6];   // [VOCAB,32]
  const float* b_out = (const float*)d_in[7];   // [VOCAB]
  const float* h0_lr = (const float*)d_in[8];
  const float* h0_rl = (const float*)d_in[9];
  const float* c0_lr = (const float*)d_in[10];
  const float* c0_rl = (const float*)d_in[11];
  float* out = (float*)d_out;

  float* xW  = (float*)d_ws;                        // [SEQ][BATCH][64]  (1 MB)
  float* hid = xW + (size_t)SEQ * BATCH * GATES;    // [SEQ][BATCH][32]  (512 KB)

  bilstm_embed_xw<<<SEQ * BATCH, 64, 0, stream>>>(idx, emb, W_ih, b_ih, b_hh, xW);
  bilstm_scan<<<2, 512, 0, stream>>>(xW, W_hh, h0_lr, h0_rl, c0_lr, c0_rl, hid);
  bilstm_logits_lsm<<<SEQ, 512, 0, stream>>>(hid, W_out, b_out, out);
}